// POSwordEmbeddingEncoderClass_29171417874568
// MI455X (gfx1250) — compile-verified
//
#include <hip/hip_runtime.h>
#include <stdint.h>

// ---------------------------------------------------------------------------
// POS+word embedding gather for MI455X (gfx1250).
//
// out[tok, 0:52]    = posMatrix[c*VOCAB + x[tok]]
// out[tok, 52:260]  = emb[x[tok]*208 : +208]
//
// 0 FLOPs, ~136 MB output stream; emb (25.4 MB) + posMatrix (6.35 MB) are
// L2-resident (192 MB L2) => pure bandwidth problem (~12 us at 23.3 TB/s).
//
// Full-TDM design:
//   pass 1: transpose posMatrix[52,V] -> posT[V,52] in workspace (6.35 MB,
//           cheap; turns the flag gather into a contiguous 208 B row read).
//   pass 2: per wave of 16 tokens (max rows for 16-bit-index gather mode):
//     - tensor_load_to_lds (gather mode) x2:
//         16 rows of posT (16x208 B) and 16 rows of emb (16x832 B) -> LDS
//     - s_wait_tensorcnt 0
//     - tensor_store_from_lds x2 with tensor_dim0_stride = 260:
//         scatters both tiles into the concatenated output rows,
//         cpol TH=NT so the 136 MB stream doesn't evict the tables.
//   All indices/descriptors are wave-uniform (readfirstlane) -> pure SALU.
//   Fallback (workspace too small): lanes gather flags directly, emb via TDM.
// ---------------------------------------------------------------------------

#define VOCAB            30522
#define NUM_POS          52
#define H_TRAIN          208
#define OUT_W            (NUM_POS + H_TRAIN)   /* 260 */
#define TOK_PER_WAVE     16                    /* gather-mode max (16b idx) */
#define WAVES_PER_BLOCK  3
#define THREADS          (WAVES_PER_BLOCK * 32)
#define FLAG_BYTES       (TOK_PER_WAVE * NUM_POS * 4)      /* 3328  */
#define EMB_BYTES        (TOK_PER_WAVE * H_TRAIN * 4)      /* 13312 */
#define LDS_PER_WAVE     (FLAG_BYTES + EMB_BYTES)          /* 16640 */
#define LDS_BYTES        (WAVES_PER_BLOCK * LDS_PER_WAVE)  /* 49920 < 64KB */
#define POS_T_BYTES      ((size_t)VOCAB * NUM_POS * 4)     /* 6.35 MB */

typedef unsigned int u32x4 __attribute__((ext_vector_type(4)));
typedef int          i32x4 __attribute__((ext_vector_type(4)));
typedef int          i32x8 __attribute__((ext_vector_type(8)));

__device__ __forceinline__ void tdm_load_to_lds(u32x4 g0, i32x8 g1, i32x4 g2, i32x4 g3) {
#if __clang_major__ >= 23
  i32x8 z8 = {0, 0, 0, 0, 0, 0, 0, 0};
  __builtin_amdgcn_tensor_load_to_lds(g0, g1, g2, g3, z8, 0);        // cpol RT
#else
  __builtin_amdgcn_tensor_load_to_lds(g0, g1, g2, g3, 0);
#endif
}

__device__ __forceinline__ void tdm_store_from_lds(u32x4 g0, i32x8 g1, i32x4 g2, i32x4 g3) {
#if __clang_major__ >= 23
  i32x8 z8 = {0, 0, 0, 0, 0, 0, 0, 0};
  __builtin_amdgcn_tensor_store_from_lds(g0, g1, g2, g3, z8, 1);     // cpol TH=NT
#else
  __builtin_amdgcn_tensor_store_from_lds(g0, g1, g2, g3, 1);
#endif
}

// Pack a D# group-0 quad: mode bits, lds byte offset, 57-bit GA, type=2.
__device__ __forceinline__ u32x4 tdm_g0(unsigned mode0, unsigned ldsOff, const void* gp) {
  const uint64_t ga = (uint64_t)(uintptr_t)gp;
  u32x4 g0;
  g0.x = mode0;
  g0.y = ldsOff;
  g0.z = (unsigned)ga;
  g0.w = (unsigned)((ga >> 32) & 0x1FFFFFFu) | 0x80000000u;          // type=2
  return g0;
}

// --------------------------- pass 1: transpose ------------------------------
__global__ void posT_transpose_kernel(const float* __restrict__ posM,
                                      float* __restrict__ posT, int total) {
  const int i = blockIdx.x * blockDim.x + threadIdx.x;
  if (i >= total) return;
  const int id = i / NUM_POS;
  const int c  = i - id * NUM_POS;
  posT[i] = posM[(size_t)c * VOCAB + id];    // coalesced write, L2-hot result
}

// --------------------------- pass 2: main gather ----------------------------
__global__ void POSwordEmbeddingEncoderClass_29171417874568_kernel(
    const int* __restrict__ x,          // [nTok] vocab ids
    const float* __restrict__ emb,      // [VOCAB, 208]
    const float* __restrict__ posM,     // [52, VOCAB]
    const float* __restrict__ posT,     // [VOCAB, 52] or nullptr
    float* __restrict__ out,            // [nTok, 260]
    int nTok) {
  extern __shared__ float smem[];

  const int wave  = __builtin_amdgcn_readfirstlane((int)(threadIdx.x >> 5));
  const int lane  = (int)(threadIdx.x & 31);
  const int gwave = (int)blockIdx.x * WAVES_PER_BLOCK + wave;   // uniform
  const int tok0  = gwave * TOK_PER_WAVE;                       // uniform
  if (tok0 >= nTok) return;                                     // uniform exit
  const int rem = nTok - tok0;
  const int cnt = rem < TOK_PER_WAVE ? rem : TOK_PER_WAVE;      // 16 in practice

  // Index fetch: fast path = four uniform 16B loads (s_load), tail = clamped.
  unsigned idx[TOK_PER_WAVE];
  if (cnt == TOK_PER_WAVE) {
    const int4* xp = (const int4*)(x + tok0);                   // 64B aligned
#pragma unroll
    for (int q = 0; q < 4; ++q) {
      const int4 a = xp[q];
      idx[q * 4 + 0] = (unsigned)a.x;
      idx[q * 4 + 1] = (unsigned)a.y;
      idx[q * 4 + 2] = (unsigned)a.z;
      idx[q * 4 + 3] = (unsigned)a.w;
    }
  } else {
#pragma unroll
    for (int j = 0; j < TOK_PER_WAVE; ++j) {
      int t = tok0 + j;
      t = t < (nTok - 1) ? t : (nTok - 1);
      idx[j] = (unsigned)x[t];
    }
  }

  // This wave's LDS slice (low 32 bits of LDS aperture addr == byte offset).
  const unsigned ldsBase =
      (unsigned)(uintptr_t)(void*)smem + (unsigned)wave * LDS_PER_WAVE;
  const unsigned ldsF = ldsBase;                 // 16 x 52 f32 flag tile
  const unsigned ldsE = ldsBase + FLAG_BYTES;    // 16 x 208 f32 emb tile

  // Row indices 0..15, two 16-bit indices per dword (VOCAB = 30522 < 2^16).
  i32x4 g2, g3;
  g2.x = (int)(idx[0]  | (idx[1]  << 16));
  g2.y = (int)(idx[2]  | (idx[3]  << 16));
  g2.z = (int)(idx[4]  | (idx[5]  << 16));
  g2.w = (int)(idx[6]  | (idx[7]  << 16));
  g3.x = (int)(idx[8]  | (idx[9]  << 16));
  g3.y = (int)(idx[10] | (idx[11] << 16));
  g3.z = (int)(idx[12] | (idx[13] << 16));
  g3.w = (int)(idx[14] | (idx[15] << 16));
  const i32x4 gz = {0, 0, 0, 0};

  const bool flagsViaTDM = (posT != nullptr);    // uniform kernarg branch

  if (flagsViaTDM) {
    // gather-load: 16 rows of posT[V,52] -> LDS
    i32x8 g1;
    g1[0] = 0x20000;                 // data_size=2 (4B), wg_mask=0
    g1[1] = (NUM_POS << 16);         // tensor_dim0 = 52
    g1[2] = (VOCAB << 16);           // tensor_dim1 = 30522 (row OOB bound)
    g1[3] = (NUM_POS << 16);         // tile_dim0 = 52
    g1[4] = cnt;                     // tile_dim1 = #valid indices
    g1[5] = NUM_POS;                 // tensor_dim0_stride = 52
    g1[6] = 0; g1[7] = 0;
    tdm_load_to_lds(tdm_g0(0x80000001u, ldsF, posT), g1, g2, g3);
  }

  {
    // gather-load: 16 rows of emb[V,208] -> LDS
    i32x8 g1;
    g1[0] = 0x20000;
    g1[1] = (H_TRAIN << 16);         // tensor_dim0 = 208
    g1[2] = (VOCAB << 16);
    g1[3] = (H_TRAIN << 16);         // tile_dim0 = 208
    g1[4] = cnt;
    g1[5] = H_TRAIN;                 // tensor_dim0_stride = 208
    g1[6] = 0; g1[7] = 0;
    tdm_load_to_lds(tdm_g0(0x80000001u, ldsE, emb), g1, g2, g3);
  }

  if (!flagsViaTDM) {
    // Fallback: lanes gather flags while the emb TDM load is in flight.
    for (int j = 0; j < cnt; ++j) {
      const unsigned id = idx[j];
      float* orow = out + (size_t)(tok0 + j) * OUT_W;
      {
        float v = posM[(size_t)lane * VOCAB + id];
        __builtin_nontemporal_store(v, orow + lane);
      }
      if (lane < NUM_POS - 32) {
        float v = posM[(size_t)(lane + 32) * VOCAB + id];
        __builtin_nontemporal_store(v, orow + lane + 32);
      }
    }
  }

  __builtin_amdgcn_s_wait_tensorcnt(0);          // gathers landed in LDS

  if (flagsViaTDM) {
    // strided store: flags tile -> out[tok, 0:52], row stride 260
    i32x8 g1;
    g1[0] = 0x20000;
    g1[1] = (NUM_POS << 16);         // tensor_dim0 = 52
    g1[2] = (cnt << 16);             // tensor_dim1 = cnt rows
    g1[3] = (NUM_POS << 16);         // tile_dim0 = 52
    g1[4] = cnt;                     // tile_dim1 = cnt
    g1[5] = OUT_W;                   // tensor_dim0_stride = 260
    g1[6] = 0; g1[7] = 0;
    tdm_store_from_lds(tdm_g0(1u, ldsF, out + (size_t)tok0 * OUT_W), g1, gz, gz);
  }

  {
    // strided store: emb tile -> out[tok, 52:260], row stride 260
    i32x8 g1;
    g1[0] = 0x20000;
    g1[1] = (H_TRAIN << 16);         // tensor_dim0 = 208
    g1[2] = (cnt << 16);             // tensor_dim1 = cnt rows
    g1[3] = (H_TRAIN << 16);         // tile_dim0 = 208
    g1[4] = cnt;                     // tile_dim1 = cnt
    g1[5] = OUT_W;                   // tensor_dim0_stride = 260
    g1[6] = 0; g1[7] = 0;
    tdm_store_from_lds(tdm_g0(1u, ldsE, out + (size_t)tok0 * OUT_W + NUM_POS),
                       g1, gz, gz);
  }
  // S_ENDPGM performs an implicit wait-idle; TDM stores drain before teardown.
}

extern "C" void kernel_launch(void* const* d_in, const int* in_sizes, int n_in,
                              void* d_out, int out_size, void* d_ws, size_t ws_size,
                              hipStream_t stream) {
  (void)n_in; (void)out_size;
  const int*   x    = (const int*)d_in[0];    // [B*S] token ids
  const float* emb  = (const float*)d_in[1];  // [VOCAB,208]
  const float* posM = (const float*)d_in[2];  // [52,VOCAB]
  float*       out  = (float*)d_out;          // [B*S,260]

  const int nTok = in_sizes[0];               // 131072

  float* posT = nullptr;
  if (d_ws != nullptr && ws_size >= POS_T_BYTES) {
    posT = (float*)d_ws;
    const int total  = VOCAB * NUM_POS;       // 1,587,144
    const int tBlk   = 256;
    const int tGrid  = (total + tBlk - 1) / tBlk;
    posT_transpose_kernel<<<tGrid, tBlk, 0, stream>>>(posM, posT, total);
  }

  const int wavesNeeded = (nTok + TOK_PER_WAVE - 1) / TOK_PER_WAVE;
  const int blocks      = (wavesNeeded + WAVES_PER_BLOCK - 1) / WAVES_PER_BLOCK;

  POSwordEmbeddingEncoderClass_29171417874568_kernel
      <<<blocks, THREADS, LDS_BYTES, stream>>>(x, emb, posM, posT, out, nTok);
}